// ExplicitSpacialEncoding_31044023615664
// MI455X (gfx1250) — compile-verified
//
#include <hip/hip_runtime.h>
#include <math.h>

typedef float v2f __attribute__((ext_vector_type(2)));
typedef float v4f __attribute__((ext_vector_type(4)));
typedef float v8f __attribute__((ext_vector_type(8)));
typedef int   v4i __attribute__((ext_vector_type(4)));

#define B_TOT   4096
#define C_CH    128
#define D_EMB   9
#define HW      64
#define KSTEPS  (HW / 4)            /* 16 wmma k-steps per channel */
#define ROW_OUT (C_CH * D_EMB)      /* 1152 */
#define LDS_STR (ROW_OUT + 1)       /* 1153 */
#define TILE_B  16
#define WAVES   16
#define THREADS (WAVES * 32)
#define CH_PER_WAVE (C_CH / WAVES)  /* 8 */
#define EPS_F   1e-10f

/* A staging: 16 rows x 64 floats, row stride 68 -> ds_load_b64 fragment reads
 * hit banks 4*row + {0..3}: all 64 banks exactly once, conflict-free. */
#define A_STR      68
#define STAGE_FL   (16 * A_STR)          /* floats per buffer = 1088 */
#define STAGE_BASE (TILE_B * LDS_STR)    /* 18448 floats */
#define SMEM_FL    (STAGE_BASE + WAVES * 2 * STAGE_FL)  /* 53264 floats = 213056 B */

#if __has_builtin(__builtin_amdgcn_global_load_async_to_lds_b128)
#define USE_ASYNC 1
#else
#define USE_ASYNC 0
#endif

/* Builtin prototype (probed via hipcc diagnostic): (int4 AS1*, int4 AS3*, Ii, Ii) */
typedef __attribute__((address_space(1))) v4i gv4i_t;
typedef __attribute__((address_space(3))) v4i lv4i_t;

__device__ __forceinline__ void copy16_to_lds(const float* g, float* l) {
#if USE_ASYNC
    /* 16B per lane, memory -> LDS direct, tracked on ASYNCcnt */
    __builtin_amdgcn_global_load_async_to_lds_b128((gv4i_t*)g, (lv4i_t*)l, 0, 0);
#else
    *(v4f*)l = *(const v4f*)g;           /* coalesced b128 + ds_store_b128 */
#endif
}

extern "C" __global__ __launch_bounds__(THREADS)
void ese_fused_kernel(const float* __restrict__ fcn,
                      const float* __restrict__ emb2,
                      float* __restrict__ out)
{
    extern __shared__ float smem[];      // [TILE_B][LDS_STR] + staging
    const int lane = threadIdx.x & 31;
    const int wave = threadIdx.x >> 5;
    const int tile = blockIdx.x;         // 0..255
    const int row  = lane & 15;          // M row / N col index
    const int hi   = lane >> 4;          // half-wave id
    const int kb   = hi * 2;             // K sub-offset (32-bit A/B operand layout)
    const int col  = row;
    const int b0   = tile * TILE_B;
    const int c0   = wave * CH_PER_WAVE;

    /* Clamp pad columns (N=9..15) to a valid address; the resulting garbage in
     * D columns 9..15 is never stored nor observed (norm reads the LDS tile). */
    const int colc = (col < D_EMB) ? col : 0;

    float* stage = smem + STAGE_BASE + wave * (2 * STAGE_FL);

    /* Stage one channel's 16x64 A tile: half-wave = one full 256B row.
     * Lane covers staged row (2j + hi), 16B chunk (lane&15). */
    auto stage_ch = [&](int c, int buf) {
        const float* g = fcn + ((size_t)(b0 + hi) * C_CH + c) * HW + row * 4;
        float*       l = stage + buf * STAGE_FL + hi * A_STR + row * 4;
#pragma unroll
        for (int j = 0; j < 8; ++j)
            copy16_to_lds(g + (size_t)j * 2 * C_CH * HW, l + j * 2 * A_STR);
    };

    stage_ch(c0, 0);                      // prologue: fill buffer 0

#pragma unroll
    for (int ci = 0; ci < CH_PER_WAVE; ++ci) {
        const int c = c0 + ci;

        if (ci + 1 < CH_PER_WAVE)
            stage_ch(c + 1, (ci + 1) & 1);     // prefetch next channel

#if USE_ASYNC
        /* in-order completion: <=8 outstanding => current buffer landed */
        if (ci + 1 < CH_PER_WAVE)
            asm volatile("s_wait_asynccnt 0x8" ::: "memory");
        else
            asm volatile("s_wait_asynccnt 0x0" ::: "memory");
#endif

        /* B fragments (tiny, L2-hot). */
        const float* Bcol = emb2 + ((size_t)c * D_EMB + colc) * HW;
        v2f breg[KSTEPS];
#pragma unroll
        for (int i = 0; i < KSTEPS; ++i)
            breg[i] = *(const v2f*)(Bcol + i * 4 + kb);

        /* A fragments from staged LDS tile (conflict-free ds_load_b64). */
        const float* abuf = stage + (ci & 1) * STAGE_FL + row * A_STR + kb;
        v2f areg[KSTEPS];
#pragma unroll
        for (int i = 0; i < KSTEPS; ++i)
            areg[i] = *(const v2f*)(abuf + i * 4);

        v8f acc = {};
#pragma unroll
        for (int i = 0; i < KSTEPS; ++i)
            acc = __builtin_amdgcn_wmma_f32_16x16x4_f32(
                      false, areg[i], false, breg[i], (short)0, acc, false, false);

        /* D: VGPR r -> M = r + hi*8, N = col; only valid columns stored. */
        if (col < D_EMB) {
            float* dst = smem + (hi * 8) * LDS_STR + c * D_EMB + col;
#pragma unroll
            for (int r = 0; r < 8; ++r)
                dst[r * LDS_STR] = acc[r];
        }
    }

    __syncthreads();

    /* Fused L2 normalization: wave w owns batch row w. */
    {
        const int r = wave;
        const float* src = smem + r * LDS_STR;
        float s = 0.0f;
        for (int j = lane; j < ROW_OUT; j += 32) {
            float v = src[j];
            s += v * v;
        }
#pragma unroll
        for (int off = 16; off >= 1; off >>= 1)
            s += __shfl_xor(s, off, 32);
        const float inv = 1.0f / sqrtf(s + EPS_F);

        float* dstrow = out + (size_t)(b0 + r) * ROW_OUT;
        for (int j = lane; j < ROW_OUT; j += 32)
            dstrow[j] = src[j] * inv;
    }
}

extern "C" void kernel_launch(void* const* d_in, const int* in_sizes, int n_in,
                              void* d_out, int out_size, void* d_ws, size_t ws_size,
                              hipStream_t stream) {
    const float* fcn  = (const float*)d_in[0];   // [4096,128,8,8] fp32
    const float* emb2 = (const float*)d_in[1];   // [1,1152,8,8]   fp32
    float* out = (float*)d_out;                  // [4096,1152]    fp32

    const size_t lds_bytes = (size_t)SMEM_FL * sizeof(float);  // 213056 B
    dim3 grid(B_TOT / TILE_B);   // 256
    dim3 block(THREADS);         // 512 = 16 wave32
    ese_fused_kernel<<<grid, block, lds_bytes, stream>>>(fcn, emb2, out);
}